// FieldAwareFactorizationMachine_55044300865787
// MI455X (gfx1250) — compile-verified
//
#include <hip/hip_runtime.h>

#define BATCH        16384
#define NUM_FIELDS   20
#define NUM_FEATURES 100000
#define LATENT_DIM   64

typedef __attribute__((ext_vector_type(2))) float v2f;
typedef __attribute__((ext_vector_type(8))) float v8f;

// One wave32 handles a tile of 16 batch rows.
//   lanes 0..15  : row = tile*16 + lane,      latent dims [0,32)
//   lanes 16..31 : row = tile*16 + (lane-16), latent dims [32,64)
// This layout makes each lane's per-field gather a contiguous 128B read, and
// it is exactly the A/B operand layout of V_WMMA_F32_16X16X4_F32 for computing
// the Gram matrix D = S * S^T (B's VGPRs == A's VGPRs for a Gram product).
// diag(D)[r] = ||s_r||^2 in full fp32 precision.
__global__ __launch_bounds__(256) void ffm_kernel(const int* __restrict__ x,
                                                  const float* __restrict__ emb,
                                                  float* __restrict__ out) {
    const unsigned tid  = blockIdx.x * blockDim.x + threadIdx.x;
    const unsigned lane = threadIdx.x & 31u;
    const unsigned wave = tid >> 5;          // tile id (16 rows per wave)
    const unsigned r    = lane & 15u;        // row within tile
    const unsigned half = lane >> 4;         // which 32 latent dims this lane owns
    const unsigned row  = wave * 16u + r;

    // ---- load this row's 20 feature indices (both lane halves load the same
    // row's indices; L0 absorbs the duplication) ----
    const int* xr = x + (size_t)row * NUM_FIELDS;
    int idxs[NUM_FIELDS];
    #pragma unroll
    for (int f = 0; f < NUM_FIELDS; ++f) idxs[f] = xr[f];

    int lin = 0;
    #pragma unroll
    for (int f = 0; f < NUM_FIELDS; ++f) lin += idxs[f];   // linear = sum of raw indices

    // ---- stream the 20 gathers; accumulate s = sum_f V_f and sumsq = sum_f ||V_f||^2 ----
    float sv[32];
    #pragma unroll
    for (int i = 0; i < 32; ++i) sv[i] = 0.0f;
    float sumsq = 0.0f;

    #pragma unroll
    for (int f = 0; f < NUM_FIELDS; ++f) {
        const float* p = emb
            + ((size_t)f * NUM_FEATURES + (size_t)idxs[f]) * LATENT_DIM
            + (size_t)half * 32u;                       // contiguous 128B per lane
        const float4* p4 = reinterpret_cast<const float4*>(p);
        #pragma unroll
        for (int c = 0; c < 8; ++c) {
            float4 v = p4[c];                           // global_load_b128
            sv[c * 4 + 0] += v.x;
            sv[c * 4 + 1] += v.y;
            sv[c * 4 + 2] += v.z;
            sv[c * 4 + 3] += v.w;
            sumsq = fmaf(v.x, v.x, sumsq);
            sumsq = fmaf(v.y, v.y, sumsq);
            sumsq = fmaf(v.z, v.z, sumsq);
            sumsq = fmaf(v.w, v.w, sumsq);
        }
    }

    // combine the two 32-dim halves of ||V||^2 for this row
    sumsq += __shfl_xor(sumsq, 16, 32);

    // ---- ||s_r||^2 for all 16 rows via fp32 WMMA Gram: D += A_c * A_c^T ----
    // Chunk c supplies K-columns {2c, 2c+1} (lanes 0-15) and {32+2c, 33+2c}
    // (lanes 16-31); 16 chunks cover all 64 latent dims.  SRC0 == SRC1.
    v8f d = {};
    #pragma unroll
    for (int c = 0; c < 16; ++c) {
        v2f a;
        a[0] = sv[2 * c + 0];
        a[1] = sv[2 * c + 1];
        d = __builtin_amdgcn_wmma_f32_16x16x4_f32(
                /*neg_a=*/false, a, /*neg_b=*/false, a,
                /*c_mod=*/(short)0, d, /*reuse_a=*/false, /*reuse_b=*/false);
    }

    // diag(D): D[v][v] lives in VGPR v, lane v;  D[v+8][v+8] in VGPR v, lane v+24.
    float acc = 0.0f;
    #pragma unroll
    for (int v = 0; v < 8; ++v) {
        float dv = d[v];
        acc += ((lane == (unsigned)v) || (lane == (unsigned)(v + 24))) ? dv : 0.0f;
    }
    // lanes 8..15 fetch their diagonal value from lanes 24..31
    float moved = __shfl(acc, (int)(lane + 16u) & 31, 32);
    float snorm = (lane < 8u) ? acc : moved;   // valid on lanes 0..15

    if (half == 0u && row < BATCH) {
        out[row] = (float)lin + 0.5f * (snorm - sumsq);
    }
}

extern "C" void kernel_launch(void* const* d_in, const int* in_sizes, int n_in,
                              void* d_out, int out_size, void* d_ws, size_t ws_size,
                              hipStream_t stream) {
    const int*   x   = (const int*)d_in[0];
    // d_in[1] = field_indices (arange, unused — field id is implicit in the loop)
    const float* emb = (const float*)d_in[2];
    float*       out = (float*)d_out;

    const int threads       = 256;                 // 8 waves per block
    const int total_waves   = BATCH / 16;          // 1024 tiles
    const int total_threads = total_waves * 32;    // 32768
    ffm_kernel<<<total_threads / threads, threads, 0, stream>>>(x, emb, out);
}